// StructureAwareVAE_90460601188767
// MI455X (gfx1250) — compile-verified
//
#include <hip/hip_runtime.h>
#include <hip/hip_bf16.h>
#include <math.h>

// ---------------------------------------------------------------------------
// Problem constants (match reference)
// ---------------------------------------------------------------------------
#define BB 2
#define NN 2048
#define DD 64
#define HH 4
#define DH 16
#define PDIM 12        // QP*3 == VP*3
#define KPACK 32       // 16 (q·k) + 12 (qp·kp) + 1 (ksq fold) + 3 pad
#define CUT2 100.0f    // CUTOFF^2
#define NEGV -1.0e9f
#define NROWS (BB*NN)          // 4096
#define PRAW_W 336             // 64*3 + 48*3
#define X2_W 112               // D + H*VP*3
#define LROW 36                // padded LDS row stride (16B aligned, conflict-free)

typedef __attribute__((ext_vector_type(2))) float v2f;
typedef __attribute__((ext_vector_type(8))) float v8f;

// V_WMMA_F32_16X16X4_F32 : D(16x16 f32) = A(16x4 f32) x B(4x16 f32) + C
static __device__ inline v8f wmma4(v2f a, v2f b, v8f c) {
  return __builtin_amdgcn_wmma_f32_16x16x4_f32(false, a, false, b, (short)0, c,
                                               false, false);
}

// Fragment layouts assumed from ISA 7.12.2 (wave32):
//  A 16x4 : lane l (m = l&15), vgpr0 = K = k0 + 2*(l>>4), vgpr1 = K+1
//  B 4x16 : lane l (n = l&15), vgpr0 = K = k0 + 2*(l>>4), vgpr1 = K+1
//  C 16x16: vgpr r, lane l -> (M = r + 8*(l>>4), N = l&15)

// ---------------------------------------------------------------------------
// CDNA5 async global->LDS staging (guarded; falls back to direct loads)
// Builtin param 0 is '__attribute__((__vector_size__(16))) int __device__ *'
// per the compiler diagnostic, i.e. non-const int4* in addrspace(1).
// ---------------------------------------------------------------------------
#if defined(__AMDGCN__) && __has_builtin(__builtin_amdgcn_global_load_async_to_lds_b128)
#define ASYNC_OK 1
typedef __attribute__((__vector_size__(4 * sizeof(int)))) int i4v;
typedef i4v __attribute__((address_space(1)))* g_b128p;
typedef i4v __attribute__((address_space(3)))* l_b128p;
// Stage one 16x32-float tile (rows padded to LROW in LDS) with 4 B128 async ops.
static __device__ inline void stage_tile(const float* g, float* l, int lane) {
  #pragma unroll
  for (int c = 0; c < 4; ++c) {
    const int q = c * 32 + lane;       // 0..127 chunks of 16B
    const int row = q >> 3, sub = q & 7;
    __builtin_amdgcn_global_load_async_to_lds_b128(
        (g_b128p)(float*)(g + row * KPACK + sub * 4),
        (l_b128p)(l + row * LROW + sub * 4), 0, 0);
  }
}
#else
#define ASYNC_OK 0
#endif

#if defined(__AMDGCN__) && __has_builtin(__builtin_amdgcn_s_wait_asynccnt)
#define WAIT_ASYNCCNT(n) __builtin_amdgcn_s_wait_asynccnt(n)
#elif defined(__AMDGCN__)
#define WAIT_ASYNCCNT(n) asm volatile("s_wait_asynccnt %0" ::"n"(n) : "memory")
#else
#define WAIT_ASYNCCNT(n)
#endif

// ---------------------------------------------------------------------------
// Kernel 1: fused projection GEMM.  Praw(4096 x 336) = X(4096x64) @
// [Wq|Wk|Wv|Wqp|Wkp|Wvp] + biases.  One wave per 16x16 output tile.
// ---------------------------------------------------------------------------
__global__ void __launch_bounds__(32)
ipa_proj_kernel(const float* __restrict__ X,
                const float* __restrict__ Wq, const float* __restrict__ bq,
                const float* __restrict__ Wk, const float* __restrict__ bk,
                const float* __restrict__ Wv, const float* __restrict__ bv,
                const float* __restrict__ Wqp, const float* __restrict__ bqp,
                const float* __restrict__ Wkp, const float* __restrict__ bkp,
                const float* __restrict__ Wvp, const float* __restrict__ bvp,
                float* __restrict__ Praw) {
  const int lane = threadIdx.x;
  const int rt = blockIdx.x;           // 256 row tiles
  const int ty = blockIdx.y;           // 21 col tiles over concat width 336
  const int row0 = rt * 16;
  const int n = lane & 15;
  const int half = lane >> 4;

  const float* W; const float* bias; int dout; int cb;
  if (ty < 4)       { W = Wq;  bias = bq;  dout = 64; cb = ty * 16; }
  else if (ty < 8)  { W = Wk;  bias = bk;  dout = 64; cb = (ty - 4) * 16; }
  else if (ty < 12) { W = Wv;  bias = bv;  dout = 64; cb = (ty - 8) * 16; }
  else if (ty < 15) { W = Wqp; bias = bqp; dout = 48; cb = (ty - 12) * 16; }
  else if (ty < 18) { W = Wkp; bias = bkp; dout = 48; cb = (ty - 15) * 16; }
  else              { W = Wvp; bias = bvp; dout = 48; cb = (ty - 18) * 16; }

  v8f acc = {};
  const float* xrow = X + (row0 + n) * DD;
  const float* wp = W + cb + n;
  #pragma unroll
  for (int k0 = 0; k0 < DD; k0 += 4) {
    const int ka = k0 + half * 2;
    v2f a, bf;
    a.x = xrow[ka];
    a.y = xrow[ka + 1];
    bf.x = wp[ka * dout];
    bf.y = wp[(ka + 1) * dout];
    acc = wmma4(a, bf, acc);
  }
  const float bv2 = bias[cb + n];
  float* po = Praw + (row0 + half * 8) * PRAW_W + ty * 16 + n;
  #pragma unroll
  for (int r = 0; r < 8; ++r) po[r * PRAW_W] = acc[r] + bv2;
}

// ---------------------------------------------------------------------------
// Kernel 2: pack pass.  Per (b,h,n):
//   Qpack[32] = [ q * softplus(w_l)/4 | (qp+c) * softplus(w_c) | 1 | 0 0 0 ]
//   Kpack[32] = [ k                   | (kp+c)                 | -0.5*wc*|kp|^2 | 0 0 0 ]
//   Vpack[32] = [ v                   | (vp+c)                 | 0 0 0 0 ]
// The row-constant -0.5*wc*|qp|^2 is softmax-invariant and dropped.
// ---------------------------------------------------------------------------
static __device__ inline float softplus_f(float x) {
  return fmaxf(x, 0.0f) + log1pf(__expf(-fabsf(x)));
}

__global__ void __launch_bounds__(256)
ipa_pack_kernel(const float* __restrict__ Praw, const float* __restrict__ coords,
                const float* __restrict__ w_l, const float* __restrict__ w_c,
                float* __restrict__ Qp, float* __restrict__ Kp,
                float* __restrict__ Vp) {
  const int idx = blockIdx.x * blockDim.x + threadIdx.x;  // 0..B*N*H-1
  if (idx >= NROWS * HH) return;
  const int h = idx & (HH - 1);
  const int row = idx >> 2;            // b*N + n
  const int b = row >> 11;             // N = 2048
  const int nn = row & (NN - 1);

  const float wl = softplus_f(w_l[h]) * 0.25f;   // folds DH^-0.5 = 1/4
  const float wc = softplus_f(w_c[h]);

  const float* pr = Praw + row * PRAW_W;
  const float* c3 = coords + row * 3;
  const float cx = c3[0], cy = c3[1], cz = c3[2];

  const int obase = ((b * HH + h) * NN + nn) * KPACK;
  float* q = Qp + obase; float* k = Kp + obase; float* v = Vp + obase;

  #pragma unroll
  for (int d = 0; d < DH; ++d) {
    q[d] = pr[h * DH + d] * wl;
    k[d] = pr[64 + h * DH + d];
    v[d] = pr[128 + h * DH + d];
  }
  float ksq = 0.0f;
  #pragma unroll
  for (int t = 0; t < PDIM; ++t) {
    const int a = t % 3;
    const float cc = (a == 0) ? cx : ((a == 1) ? cy : cz);
    const float qp = pr[192 + h * PDIM + t] + cc;
    const float kp = pr[240 + h * PDIM + t] + cc;
    const float vp = pr[288 + h * PDIM + t] + cc;
    q[16 + t] = qp * wc;
    k[16 + t] = kp;
    v[16 + t] = vp;
    ksq += kp * kp;
  }
  q[28] = 1.0f; q[29] = 0.0f; q[30] = 0.0f; q[31] = 0.0f;
  k[28] = -0.5f * wc * ksq; k[29] = 0.0f; k[30] = 0.0f; k[31] = 0.0f;
  v[28] = 0.0f; v[29] = 0.0f; v[30] = 0.0f; v[31] = 0.0f;
}

// ---------------------------------------------------------------------------
// Kernel 3: flash attention.  One wave per (b,h, 16-row query tile).
// Double-buffers K/V tiles in LDS via async global->LDS B128 copies
// (s_wait_asynccnt <= 8 keeps exactly the next tile in flight), then per
// tile: 8 WMMAs (logits, K=32) + cutoff mask + online softmax + LDS
// transpose of P + 8 WMMAs (P@Vpack).  Writes into concat layout X2.
// ---------------------------------------------------------------------------
__global__ void __launch_bounds__(32)
ipa_flash_kernel(const float* __restrict__ Qp, const float* __restrict__ Kp,
                 const float* __restrict__ Vp, const float* __restrict__ coords,
                 float* __restrict__ X2) {
  const int lane = threadIdx.x;
  const int bh = blockIdx.x;           // 0..7
  const int it = blockIdx.y;           // 0..127
  const int b = bh >> 2, h = bh & 3;
  const int i0 = it * 16;
  const int n = lane & 15;
  const int half = lane >> 4;
  const int mhi = half * 8;

  const float* qb = Qp + bh * (NN * KPACK);
  const float* kb = Kp + bh * (NN * KPACK);
  const float* vb = Vp + bh * (NN * KPACK);
  const float* cb = coords + b * (NN * 3);

  // Q A-fragments (16 rows x K=32) resident in registers.
  v2f qa[8];
  const float* qrow = qb + (i0 + n) * KPACK;
  #pragma unroll
  for (int s = 0; s < 8; ++s) {
    const int ka = 4 * s + half * 2;
    qa[s].x = qrow[ka];
    qa[s].y = qrow[ka + 1];
  }
  // Query coords for this lane's 8 output rows.
  float cix[8], ciy[8], ciz[8];
  #pragma unroll
  for (int r = 0; r < 8; ++r) {
    const float* c = cb + (i0 + mhi + r) * 3;
    cix[r] = c[0]; ciy[r] = c[1]; ciz[r] = c[2];
  }

  v8f o0 = {}, o1 = {};
  float rmax[8], rsum[8];
  #pragma unroll
  for (int r = 0; r < 8; ++r) { rmax[r] = -3.0e38f; rsum[r] = 0.0f; }

#if ASYNC_OK
  __shared__ float kbuf[2][16 * LROW];
  __shared__ float vbuf[2][16 * LROW];
#endif
  __shared__ float pt[16 * 17];        // padded P-transpose staging

#if ASYNC_OK
  stage_tile(kb, &kbuf[0][0], lane);   // prefetch tile 0
  stage_tile(vb, &vbuf[0][0], lane);
#endif
  int buf = 0;

  for (int jt = 0; jt < NN / 16; ++jt) {
    const int j0 = jt * 16;
    v8f s = {};

#if ASYNC_OK
    if (jt + 1 < NN / 16) {            // prefetch next tile into other buffer
      stage_tile(kb + (j0 + 16) * KPACK, &kbuf[buf ^ 1][0], lane);
      stage_tile(vb + (j0 + 16) * KPACK, &vbuf[buf ^ 1][0], lane);
      WAIT_ASYNCCNT(8);                // current tile's 8 copies are done
    } else {
      WAIT_ASYNCCNT(0);
    }
    const float* kl = &kbuf[buf][0];
    const float* vl = &vbuf[buf][0];

    // ---- logits tile S = Qpack @ Kpack^T (K=32), B-frags from LDS ----
    #pragma unroll
    for (int st = 0; st < 8; ++st) {
      const int ka = 4 * st + half * 2;
      v2f bf;
      bf.x = kl[n * LROW + ka];
      bf.y = kl[n * LROW + ka + 1];
      s = wmma4(qa[st], bf, s);
    }
#else
    const float* krow = kb + (j0 + n) * KPACK;
    #pragma unroll
    for (int st = 0; st < 8; ++st) {
      const int ka = 4 * st + half * 2;
      v2f bf;
      bf.x = krow[ka];
      bf.y = krow[ka + 1];
      s = wmma4(qa[st], bf, s);
    }
#endif

    // ---- cutoff mask (replace with NEG like the reference) ----
    const float* cj = cb + (j0 + n) * 3;
    const float cjx = cj[0], cjy = cj[1], cjz = cj[2];
    #pragma unroll
    for (int r = 0; r < 8; ++r) {
      const float dx = cix[r] - cjx, dy = ciy[r] - cjy, dz = ciz[r] - cjz;
      const float d2 = dx * dx + dy * dy + dz * dz;
      if (d2 > CUT2) s[r] = NEGV;
    }

    // ---- online softmax (row stats shared across the 16-lane half) ----
    float osc[8];
    #pragma unroll
    for (int r = 0; r < 8; ++r) {
      float v = s[r];
      v = fmaxf(v, __shfl_xor(v, 1, 32));
      v = fmaxf(v, __shfl_xor(v, 2, 32));
      v = fmaxf(v, __shfl_xor(v, 4, 32));
      v = fmaxf(v, __shfl_xor(v, 8, 32));
      const float nm = fmaxf(rmax[r], v);
      const float sc = __expf(rmax[r] - nm);
      rmax[r] = nm;
      const float p = __expf(s[r] - nm);
      s[r] = p;
      float ps = p;
      ps += __shfl_xor(ps, 1, 32);
      ps += __shfl_xor(ps, 2, 32);
      ps += __shfl_xor(ps, 4, 32);
      ps += __shfl_xor(ps, 8, 32);
      rsum[r] = rsum[r] * sc + ps;
      osc[r] = sc;
    }
    #pragma unroll
    for (int r = 0; r < 8; ++r) { o0[r] *= osc[r]; o1[r] *= osc[r]; }

    // ---- transpose P (C-layout -> A-layout) through LDS ----
    __syncthreads();                   // protect previous iteration's reads
    #pragma unroll
    for (int r = 0; r < 8; ++r) pt[(mhi + r) * 17 + n] = s[r];
    __syncthreads();

    // ---- O += P @ Vpack tile (K=16, two 16-col output tiles) ----
    #pragma unroll
    for (int st = 0; st < 4; ++st) {
      const int ka = 4 * st + half * 2;
      v2f pa, bf0, bf1;
      pa.x = pt[n * 17 + ka];
      pa.y = pt[n * 17 + ka + 1];
#if ASYNC_OK
      bf0.x = vl[ka * LROW + n];
      bf0.y = vl[(ka + 1) * LROW + n];
      bf1.x = vl[ka * LROW + 16 + n];
      bf1.y = vl[(ka + 1) * LROW + 16 + n];
#else
      const float* vbase = vb + j0 * KPACK;
      bf0.x = vbase[ka * KPACK + n];
      bf0.y = vbase[(ka + 1) * KPACK + n];
      bf1.x = vbase[ka * KPACK + 16 + n];
      bf1.y = vbase[(ka + 1) * KPACK + 16 + n];
#endif
      o0 = wmma4(pa, bf0, o0);
      o1 = wmma4(pa, bf1, o1);
    }
    buf ^= 1;
  }

  // ---- normalize and scatter into concat layout X2(row, 112) ----
  const int rowbase = b * NN + i0 + mhi;
  #pragma unroll
  for (int r = 0; r < 8; ++r) {
    const float inv = 1.0f / rsum[r];
    const int row = (rowbase + r) * X2_W;
    X2[row + h * DH + n] = o0[r] * inv;                       // out_linear
    if (n < PDIM) X2[row + DD + h * PDIM + n] = o1[r] * inv;  // out_points
  }
}

// ---------------------------------------------------------------------------
// Kernel 4: output projection.  out(4096x64) = X2(4096x112) @ Wo + bo.
// ---------------------------------------------------------------------------
__global__ void __launch_bounds__(32)
ipa_outproj_kernel(const float* __restrict__ X2, const float* __restrict__ Wo,
                   const float* __restrict__ bo, float* __restrict__ out) {
  const int lane = threadIdx.x;
  const int rt = blockIdx.x;           // 256
  const int ct = blockIdx.y;           // 4
  const int row0 = rt * 16, c0 = ct * 16;
  const int n = lane & 15;
  const int half = lane >> 4;

  v8f acc = {};
  const float* xrow = X2 + (row0 + n) * X2_W;
  const float* wp = Wo + c0 + n;
  #pragma unroll
  for (int k0 = 0; k0 < X2_W; k0 += 4) {
    const int ka = k0 + half * 2;
    v2f a, bf;
    a.x = xrow[ka];
    a.y = xrow[ka + 1];
    bf.x = wp[ka * DD];
    bf.y = wp[(ka + 1) * DD];
    acc = wmma4(a, bf, acc);
  }
  const float bv = bo[c0 + n];
  float* po = out + (row0 + half * 8) * DD + c0 + n;
  #pragma unroll
  for (int r = 0; r < 8; ++r) po[r * DD] = acc[r] + bv;
}

// ---------------------------------------------------------------------------
// Host launcher
// ---------------------------------------------------------------------------
extern "C" void kernel_launch(void* const* d_in, const int* in_sizes, int n_in,
                              void* d_out, int out_size, void* d_ws, size_t ws_size,
                              hipStream_t stream) {
  const float* features = (const float*)d_in[0];
  const float* coords   = (const float*)d_in[1];
  const float* Wq  = (const float*)d_in[2];  const float* bq  = (const float*)d_in[3];
  const float* Wk  = (const float*)d_in[4];  const float* bk  = (const float*)d_in[5];
  const float* Wv  = (const float*)d_in[6];  const float* bv  = (const float*)d_in[7];
  const float* Wqp = (const float*)d_in[8];  const float* bqp = (const float*)d_in[9];
  const float* Wkp = (const float*)d_in[10]; const float* bkp = (const float*)d_in[11];
  const float* Wvp = (const float*)d_in[12]; const float* bvp = (const float*)d_in[13];
  const float* Wo  = (const float*)d_in[14]; const float* bo  = (const float*)d_in[15];
  const float* w_c = (const float*)d_in[16];
  const float* w_l = (const float*)d_in[17];
  float* out = (float*)d_out;

  float* ws = (float*)d_ws;
  // Workspace map (floats):
  //   Praw : [0, 4096*336)                raw projections; dead after pack
  //   Qp/Kp/Vp : 3 x B*H*N*32             packed fused-logit operands
  //   X2   : [0, 4096*112) (reuses Praw)  concat attention output
  float* Praw = ws;
  float* Qp = ws + (size_t)NROWS * PRAW_W;
  float* Kp = Qp + (size_t)BB * HH * NN * KPACK;
  float* Vp = Kp + (size_t)BB * HH * NN * KPACK;
  float* X2 = ws;

  ipa_proj_kernel<<<dim3(NROWS / 16, 21), 32, 0, stream>>>(
      features, Wq, bq, Wk, bk, Wv, bv, Wqp, bqp, Wkp, bkp, Wvp, bvp, Praw);

  ipa_pack_kernel<<<dim3((NROWS * HH) / 256), 256, 0, stream>>>(
      Praw, coords, w_l, w_c, Qp, Kp, Vp);

  ipa_flash_kernel<<<dim3(BB * HH, NN / 16), 32, 0, stream>>>(
      Qp, Kp, Vp, coords, X2);

  ipa_outproj_kernel<<<dim3(NROWS / 16, DD / 16), 32, 0, stream>>>(
      X2, Wo, bo, out);
}